// HierPostProcessor_76407468195900
// MI455X (gfx1250) — compile-verified
//
#include <hip/hip_runtime.h>

// ---------------------------------------------------------------------------
// HierPostProcessor for MI455X (gfx1250)
// Streaming sigmoid*sigmoid argmax over 784 positions per (ROI, class),
// fed by CDNA5 async global->LDS copies (ASYNCcnt pipeline), wave32 butterfly
// reduction, scalar box transform on lane 0.
// ---------------------------------------------------------------------------

#define NCLS 80
#define MRES 28
#define HW (MRES * MRES)          // 784
#define WAVES_PER_BLK 8
#define CLS_PER_WAVE (NCLS / WAVES_PER_BLK)   // 10
#define IMG_W 1024
#define IMG_H 800

// ---- CDNA5 async global->LDS helpers --------------------------------------
#if defined(__has_builtin)
#if __has_builtin(__builtin_amdgcn_global_load_async_to_lds_b128)
#define HAVE_ASYNC_B128 1
#endif
#if __has_builtin(__builtin_amdgcn_s_wait_asynccnt)
#define HAVE_WAIT_ASYNC 1
#endif
#endif

typedef int v4i __attribute__((vector_size(16)));
typedef __attribute__((address_space(1))) v4i gv4i;   // global v4i
typedef __attribute__((address_space(3))) v4i lv4i;   // LDS v4i

__device__ __forceinline__ void async_g2lds_b128(const float* g, float* l) {
#if defined(HAVE_ASYNC_B128)
  __builtin_amdgcn_global_load_async_to_lds_b128(
      (gv4i*)g, (lv4i*)l, /*offset=*/0, /*cpol=*/0);
#else
  unsigned lds32 = (unsigned)(unsigned long long)(lv4i*)l;
  unsigned long long ga = (unsigned long long)g;
  asm volatile("global_load_async_to_lds_b128 %0, %1, off"
               :: "v"(lds32), "v"(ga)
               : "memory");
#endif
}

#if defined(HAVE_WAIT_ASYNC)
#define WAIT_ASYNC(n) __builtin_amdgcn_s_wait_asynccnt(n)
#else
#define WAIT_ASYNC(n) asm volatile("s_wait_asynccnt %0" ::"n"(n) : "memory")
#endif

__device__ __forceinline__ float fast_sigmoid(float x) {
  // sigmoid(x) = 1 / (1 + 2^(-x*log2(e)))  -> v_exp_f32 + v_rcp_f32
  return __builtin_amdgcn_rcpf(1.0f +
                               __builtin_amdgcn_exp2f(x * -1.44269504088896f));
}

// Issue the 7 async b128 transfers that move one 784-float class row to LDS.
__device__ __forceinline__ void issue_row(const float* src, float* dst,
                                          int lane) {
#pragma unroll
  for (int i = 0; i < 7; ++i) {
    int e = i * 128 + lane * 4;      // 32 lanes x 4 floats per issue
    if (e < HW) {                    // tail issue: lanes 0..3 only (16 floats)
      async_g2lds_b128(src + e, dst + e);
    }
  }
}

__global__ __launch_bounds__(256, 2) void hier_post_kernel(
    const float* __restrict__ cls,      // [T, C, M, M]
    const float* __restrict__ reg,      // [T, 4, M, M]
    const float* __restrict__ ctr,      // [T, 1, M, M]
    const float* __restrict__ boxes,    // [T, 4]
    const float* __restrict__ bscores,  // [T]
    float* __restrict__ out,            // [T, C, 5]
    float* __restrict__ labels,         // [C]
    float* __restrict__ keep)           // [T, C]
{
  __shared__ float ctr_s[HW];
  __shared__ float stage[WAVES_PER_BLK][2][HW];

  const int t    = blockIdx.x;
  const int tid  = threadIdx.x;
  const int wave = tid >> 5;
  const int lane = tid & 31;

  const float* clsw = cls + ((size_t)t * NCLS + (size_t)wave * CLS_PER_WAVE) * HW;

  // Prefetch this wave's first class row while we stage centerness.
  issue_row(clsw, &stage[wave][0][0], lane);

  // sigmoid(centerness) -> LDS, shared by all waves of this block.
  const float* ctr_t = ctr + (size_t)t * HW;
  for (int i = tid; i < HW; i += 256) {
    ctr_s[i] = fast_sigmoid(ctr_t[i]);
  }
  __syncthreads();

  const float bx = boxes[t * 4 + 0];
  const float by = boxes[t * 4 + 1];
  const float bw = boxes[t * 4 + 2] - bx;
  const float bh = boxes[t * 4 + 3] - by;
  const float bs = bscores[t];

  for (int j = 0; j < CLS_PER_WAVE; ++j) {
    const int sel = j & 1;
    if (j + 1 < CLS_PER_WAVE) {
      // Prefetch next class row into the other buffer, then wait until only
      // those 7 transfers remain in flight (async ops complete in order, so
      // the current buffer's 7 transfers are done).
      issue_row(clsw + (size_t)(j + 1) * HW, &stage[wave][sel ^ 1][0], lane);
      WAIT_ASYNC(7);
    } else {
      WAIT_ASYNC(0);
    }
    asm volatile("" ::: "memory");   // keep LDS reads below the wait

    // Per-lane strided scan of 784 LDS values: sigmoid(cls)*ctr_s, argmax.
    const float* buf = &stage[wave][sel][0];
    float best = -3.402823466e38f;
    int   bidx = 0;
    for (int k = lane; k < HW; k += 32) {
      float v = ctr_s[k] * fast_sigmoid(buf[k]);
      if (v > best) { best = v; bidx = k; }
    }
    // Wave32 butterfly argmax (first-occurrence tie-break like jnp.argmax).
#pragma unroll
    for (int off = 16; off >= 1; off >>= 1) {
      float ov = __shfl_xor(best, off, 32);
      int   oi = __shfl_xor(bidx, off, 32);
      if (ov > best || (ov == best && oi < bidx)) { best = ov; bidx = oi; }
    }

    if (lane == 0) {
      const int c = wave * CLS_PER_WAVE + j;
      const float* regt = reg + (size_t)t * 4 * HW;
      const float rl = regt[0 * HW + bidx];
      const float rt = regt[1 * HW + bidx];
      const float rr = regt[2 * HW + bidx];
      const float rb = regt[3 * HW + bidx];
      // hw index = i*M + j  ->  loc = (j + 0.5, i + 0.5)
      const float lx = (float)(bidx % MRES) + 0.5f;
      const float ly = (float)(bidx / MRES) + 0.5f;
      const float fm = (float)MRES;
      float x1 = (lx - rl) / fm * bw + bx;
      float y1 = (ly - rt) / fm * bh + by;
      float x2 = (lx + rr) / fm * bw + bx;
      float y2 = (ly + rb) / fm * bh + by;
      x1 = fminf(fmaxf(x1, 0.0f), (float)(IMG_W - 1));
      y1 = fminf(fmaxf(y1, 0.0f), (float)(IMG_H - 1));
      x2 = fminf(fmaxf(x2, 0.0f), (float)(IMG_W - 1));
      y2 = fminf(fmaxf(y2, 0.0f), (float)(IMG_H - 1));
      const float score = sqrtf(sqrtf(best) * bs);

      const size_t o = ((size_t)t * NCLS + c) * 5;
      out[o + 0] = x1;
      out[o + 1] = y1;
      out[o + 2] = x2;
      out[o + 3] = y2;
      out[o + 4] = score;
      keep[(size_t)t * NCLS + c] =
          (((x2 - x1 + 1.0f) >= 0.0f) && ((y2 - y1 + 1.0f) >= 0.0f)) ? 1.0f
                                                                     : 0.0f;
    }
  }

  if (t == 0 && tid < NCLS) {
    labels[tid] = (float)(2 + tid);   // torch.range(2, 1+C)
  }
}

extern "C" void kernel_launch(void* const* d_in, const int* in_sizes, int n_in,
                              void* d_out, int out_size, void* d_ws,
                              size_t ws_size, hipStream_t stream) {
  (void)n_in; (void)out_size; (void)d_ws; (void)ws_size;
  const float* cls    = (const float*)d_in[0];
  const float* reg    = (const float*)d_in[1];
  const float* ctr    = (const float*)d_in[2];
  const float* boxes  = (const float*)d_in[3];
  const float* bsc    = (const float*)d_in[4];
  const int T = in_sizes[4];  // boxes_scores has T elements

  float* out    = (float*)d_out;                   // [T, C, 5]
  float* labels = out + (size_t)T * NCLS * 5;      // [C]
  float* keep   = labels + NCLS;                   // [T, C]

  hipLaunchKernelGGL(hier_post_kernel, dim3(T), dim3(256), 0, stream,
                     cls, reg, ctr, boxes, bsc, out, labels, keep);
}